// DifferentiableOptics_36017595744674
// MI455X (gfx1250) — compile-verified
//
#include <hip/hip_runtime.h>
#include <hip/hip_bf16.h>
#include <math.h>
#include <stdint.h>

// ---------------------------------------------------------------------------
// DifferentiableOptics PSF pipeline for gfx1250 (MI455X, wave32, WMMA + async)
//
// Constants from the reference:
//   G=256, B=128, NZ=15, FOV=32, F_M=0.025, F_NUMBER=2, PUPIL_DIAM=0.0125,
//   PIXEL_SIZE=3.45e-6, defocus const = F_M^2/(8*F_N^2) = 1.953125e-5
//
// fftshift(FFT2(ifftshift(p))) == (-1)^(u+v) * FFT2(p * (-1)^(x+y)) for even
// sizes; the output sign vanishes under |.|^2, so only the input checkerboard
// sign is applied.
// ---------------------------------------------------------------------------

#define G      256
#define GG     65536
#define BATCH  128
#define NZ     15
#define FOV    32
#define CHUNK  16                // batches per pipeline iteration (one WMMA M-tile)
#define IMGS   (CHUNK * 3)       // images in flight per iteration

typedef float v2f __attribute__((ext_vector_type(2)));
typedef float v8f __attribute__((ext_vector_type(8)));

// Truncating a generic pointer to __shared__ data yields the LDS byte offset
// (flat LDS aperture keeps the offset in addr[31:0], per ISA 10.2).
__device__ __forceinline__ uint32_t lds_off(const void* p)
{
    return (uint32_t)(uintptr_t)p;
}

// ---------------------------------------------------------------------------
// K1: opd[m][p] = sum_k coeffs[m][k] * basis[k][p]  via V_WMMA_F32_16X16X4_F32
//     coeffs[m][k] = zern[k] + (k==3)*defocus_wfe(d_obj[base+m], focus)
//     One wave per 16x16 tile; K=15 padded to 16 -> 4 WMMA steps of K=4.
//     K-padding is branchless (clamped index + masked value) so EXEC stays
//     all-ones around the WMMAs.
// ---------------------------------------------------------------------------
__global__ void __launch_bounds__(128)
gemm_opd_wmma(float* __restrict__ opd, const float* __restrict__ basis,
              const float* __restrict__ zern, const float* __restrict__ d_obj,
              const float* __restrict__ focus, int base)
{
    const int lane = threadIdx.x & 31;
    const int wave = threadIdx.x >> 5;
    const int ptile = (blockIdx.x * 4 + wave) * 16;   // pixel tile base
    const int h  = lane >> 4;                          // lane half (0/1)
    const int ln = lane & 15;

    // Defocus for this lane's A-rows (A-matrix: lanes 0-15 carry M=ln; the
    // upper half carries the same M rows for K=2,3).
    const float fd  = focus[0];
    const float dob = d_obj[base + ln];
    const float dwfe = (0.025f * 0.025f / 32.0f) * (1.0f / fd - 1.0f / (dob + 1e-8f));

    v8f acc = {0.f, 0.f, 0.f, 0.f, 0.f, 0.f, 0.f, 0.f};

#pragma unroll
    for (int s = 0; s < 4; ++s) {
        const int k0 = 4 * s + 2 * h;    // this lane's K indices for this step
        const int k1 = k0 + 1;
        const int k0c = (k0 < NZ) ? k0 : 0;
        const int k1c = (k1 < NZ) ? k1 : 0;
        const float m0 = (k0 < NZ) ? 1.f : 0.f;
        const float m1 = (k1 < NZ) ? 1.f : 0.f;

        // A fragment (16x4 f32): lane%16 = M, {VGPR, lane-half} = K
        float a0 = zern[k0c] * m0;
        float a1 = zern[k1c] * m1;
        if (k0 == 3) a0 += dwfe;
        if (k1 == 3) a1 += dwfe;
        v2f a; a.x = a0; a.y = a1;

        // B fragment (4x16 f32), mirrored layout: lane%16 = N, {VGPR, half} = K
        v2f b;
        b.x = basis[k0c * GG + ptile + ln] * m0;
        b.y = basis[k1c * GG + ptile + ln] * m1;

        if (s < 3) {   // stream-prefetch next K-slab of the basis
            __builtin_prefetch(&basis[(k0 + 4) * GG + ptile + ln], 0, 0);
        }

        // D = A*B + C   (8 args: neg_a, A, neg_b, B, c_mod, C, reuse_a, reuse_b)
        acc = __builtin_amdgcn_wmma_f32_16x16x4_f32(
                  false, a, false, b, (short)0, acc, false, false);
    }

    // C/D layout: VGPR g, lane half hh -> M = g + 8*hh, N = lane%16
#pragma unroll
    for (int g = 0; g < 8; ++g)
        opd[(g + 8 * h) * GG + ptile + ln] = acc[g];
}

// ---------------------------------------------------------------------------
// K2: rows: pupil = sign * aperture * e^{i*2*pi*opd/lambda}; 256-pt radix-2
//     DIF FFT per row in LDS with an LDS twiddle table (all twiddles are
//     powers of W_256). Output is bit-reverse-scattered inside LDS, then the
//     contiguous 2KB row is streamed out via GLOBAL_STORE_ASYNC_FROM_LDS_B128
//     overlapped with the next row's FFT (ASYNCcnt drained just before row2
//     is reused).
//     grid = (8 row-groups, IMGS images), block = 256.
// ---------------------------------------------------------------------------
__global__ void __launch_bounds__(256)
rowfft(float2* __restrict__ field, const float* __restrict__ opd,
       const float* __restrict__ aperture, const float* __restrict__ wl)
{
    __shared__ float2 row[G];
    __shared__ float2 row2[G];
    __shared__ float2 tw[G / 2];

    const int img = blockIdx.y;            // 0..IMGS-1
    const int m = img / 3, ch = img - 3 * m;
    const float w2pi = 6.28318530717958647692f / wl[ch];
    const int x = threadIdx.x;
    const int rx = (int)(__brev((unsigned)x) >> 24);

    if (x < G / 2) {                       // tw[t] = W_256^t = e^{-2*pi*i*t/256}
        float sn, cs;
        __sincosf(-6.28318530717958647692f * (float)x / 256.0f, &sn, &cs);
        tw[x] = make_float2(cs, sn);
    }
    __syncthreads();

    for (int r = 0; r < 32; ++r) {
        const int y = blockIdx.x * 32 + r;
        const float ap = aperture[y * G + x];
        float2 v = make_float2(0.f, 0.f);
        if (ap != 0.f) {
            const float sgn = ((x ^ y) & 1) ? -1.f : 1.f;
            float sn, cs;
            sincosf(opd[m * GG + y * G + x] * w2pi, &sn, &cs);
            v = make_float2(ap * cs * sgn, ap * sn * sgn);
        }
        row[x] = v;
        __syncthreads();

        int sh = 0;
        for (int hh = 128; hh >= 1; hh >>= 1, ++sh) {
            if ((x & hh) == 0) {
                const int j = x & (hh - 1);
                const float2 pa = row[x];
                const float2 pb = row[x + hh];
                const float2 w  = tw[j << sh];          // e^{-i*pi*j/hh}
                const float dr = pa.x - pb.x, di = pa.y - pb.y;
                row[x]      = make_float2(pa.x + pb.x, pa.y + pb.y);
                row[x + hh] = make_float2(dr * w.x - di * w.y, dr * w.y + di * w.x);
            }
            __syncthreads();
        }

        // Drain the previous row's async stores before overwriting row2.
        asm volatile("s_wait_asynccnt 0x0" ::: "memory");
        __syncthreads();
        row2[rx] = row[x];                  // bit-reversal as LDS scatter
        __syncthreads();

        if (x < 128) {                      // 128 lanes x 16B = one 2KB row
            const uint32_t lds = lds_off(&row2[2 * x]);
            const unsigned long long ga =
                (unsigned long long)(uintptr_t)(field + (size_t)img * GG + y * G + 2 * x);
            asm volatile("global_store_async_from_lds_b128 %0, %1, off"
                         :: "v"(ga), "v"(lds) : "memory");
        }
    }
    // S_ENDPGM performs an implicit wait-idle, so outstanding async stores
    // complete before the next kernel in the stream observes `field`.
}

// ---------------------------------------------------------------------------
// K3: columns: fill a 64KB 256x32 complex LDS tile directly from memory via
//     GLOBAL_LOAD_ASYNC_TO_LDS_B128 (two contiguous 256B segments per wave
//     issue), then radix-2 DIF along y with all 256 threads doing 16
//     butterflies/stage; write |field|^2 with bit-reversed rows.
//     grid = (8 col-groups, IMGS), block = 256.
// ---------------------------------------------------------------------------
__global__ void __launch_bounds__(256)
colfft(float* __restrict__ psf, const float2* __restrict__ field)
{
    __shared__ float2 tile[G][32];
    const int img = blockIdx.y;
    const int c0  = blockIdx.x * 32;
    const int tx  = threadIdx.x & 31;
    const int ty  = threadIdx.x >> 5;

    const float2* fimg = field + (size_t)img * GG;

    {   // async tile fill: 16 passes x 256 lanes x 16B = 64KB
        const int l16 = threadIdx.x & 15;       // 16B chunk within a row
        const int yq  = threadIdx.x >> 4;       // row within a pass (0..15)
#pragma unroll 4
        for (int p = 0; p < 16; ++p) {
            const int y = p * 16 + yq;
            const uint32_t lds = lds_off(&tile[y][l16 * 2]);
            const unsigned long long ga =
                (unsigned long long)(uintptr_t)(fimg + y * G + c0 + l16 * 2);
            asm volatile("global_load_async_to_lds_b128 %0, %1, off"
                         :: "v"(lds), "v"(ga) : "memory");
        }
        asm volatile("s_wait_asynccnt 0x0" ::: "memory");
    }
    __syncthreads();

    for (int hh = 128; hh >= 1; hh >>= 1) {
        for (int k = 0; k < 16; ++k) {
            const int id   = threadIdx.x + 256 * k;   // 4096 disjoint butterflies
            const int col  = id & 31;
            const int j128 = id >> 5;
            const int j    = j128 & (hh - 1);
            const int i    = ((j128 & ~(hh - 1)) << 1) | j;
            const float2 pa = tile[i][col];
            const float2 pb = tile[i + hh][col];
            float ws, wc;
            __sincosf(-3.14159265358979323846f * (float)j / (float)hh, &ws, &wc);
            const float dr = pa.x - pb.x, di = pa.y - pb.y;
            tile[i][col]      = make_float2(pa.x + pb.x, pa.y + pb.y);
            tile[i + hh][col] = make_float2(dr * wc - di * ws, dr * ws + di * wc);
        }
        __syncthreads();
    }

    float* pimg = psf + (size_t)img * GG;
#pragma unroll 4
    for (int i = 0; i < 32; ++i) {
        const int y  = i * 8 + ty;
        const int ry = (int)(__brev((unsigned)y) >> 24);
        const float2 v = tile[y][tx];
        pimg[ry * G + c0 + tx] = v.x * v.x + v.y * v.y;
    }
}

// ---------------------------------------------------------------------------
// K4: bilinear grid-sample (align_corners=False, zero padding) + normalize.
//     grid = IMGS, block = 256 (4 output pixels/thread).
// ---------------------------------------------------------------------------
__device__ __forceinline__ float bil_tap(const float* __restrict__ img,
                                         float xf, float yf, float w)
{
    const bool valid = (xf >= 0.f) & (xf <= (float)(G - 1)) &
                       (yf >= 0.f) & (yf <= (float)(G - 1));
    const int xi = (int)fminf(fmaxf(xf, 0.f), (float)(G - 1));
    const int yi = (int)fminf(fmaxf(yf, 0.f), (float)(G - 1));
    return valid ? img[yi * G + xi] * w : 0.f;
}

__global__ void __launch_bounds__(256)
sample_norm(float* __restrict__ out, const float* __restrict__ psf,
            const float* __restrict__ wl, int arm, int base)
{
    __shared__ float red[256];
    const int img = blockIdx.x;
    const int m = img / 3, ch = img - 3 * m;
    const float lam  = wl[ch];
    // zoom = PIXEL_SIZE*FOV*PUPIL_DIAM / (lambda*F_M*G)
    const float zoom = (3.45e-6f * 32.0f * 0.0125f) / (lam * 0.025f * 256.0f);
    const float* pimg = psf + (size_t)img * GG;

    float vals[4];
    float psum = 0.f;
#pragma unroll
    for (int q = 0; q < 4; ++q) {
        const int s  = threadIdx.x + 256 * q;
        const int oy = s >> 5, ox = s & 31;
        const float gx = zoom * (2.0f * (float)ox / 31.0f - 1.0f);
        const float gy = zoom * (2.0f * (float)oy / 31.0f - 1.0f);
        const float xx = ((gx + 1.0f) * (float)G - 1.0f) * 0.5f;
        const float yy = ((gy + 1.0f) * (float)G - 1.0f) * 0.5f;
        const float x0 = floorf(xx), y0 = floorf(yy);
        const float tx = xx - x0,  ty = yy - y0;
        float v = bil_tap(pimg, x0,       y0,       (1.f - tx) * (1.f - ty))
                + bil_tap(pimg, x0 + 1.f, y0,       tx * (1.f - ty))
                + bil_tap(pimg, x0,       y0 + 1.f, (1.f - tx) * ty)
                + bil_tap(pimg, x0 + 1.f, y0 + 1.f, tx * ty);
        vals[q] = v;
        psum += v;
    }

    red[threadIdx.x] = psum;
    __syncthreads();
    for (int o = 128; o >= 1; o >>= 1) {
        if (threadIdx.x < (unsigned)o) red[threadIdx.x] += red[threadIdx.x + o];
        __syncthreads();
    }
    const float inv = 1.0f / (red[0] + 1e-8f);

    const size_t obase = ((size_t)arm * BATCH * 3 + (size_t)(base + m) * 3 + ch) * 1024;
#pragma unroll
    for (int q = 0; q < 4; ++q)
        out[obase + threadIdx.x + 256 * q] = vals[q] * inv;
}

// ---------------------------------------------------------------------------
// Host orchestration: 2 arms x 8 chunks of 16 batches; workspace ~40 MB:
//   opd   : CHUNK*GG floats          (4 MB)
//   field : IMGS*GG  float2          (24 MB)
//   psf   : IMGS*GG  floats          (12 MB)
// ---------------------------------------------------------------------------
extern "C" void kernel_launch(void* const* d_in, const int* in_sizes, int n_in,
                              void* d_out, int out_size, void* d_ws, size_t ws_size,
                              hipStream_t stream)
{
    (void)in_sizes; (void)n_in; (void)out_size; (void)ws_size;

    const float* d_obj  = (const float*)d_in[0];
    const float* foc0   = (const float*)d_in[1];
    const float* foc90  = (const float*)d_in[2];
    const float* zern0  = (const float*)d_in[3];
    const float* zern90 = (const float*)d_in[4];
    const float* basis  = (const float*)d_in[5];
    const float* aper   = (const float*)d_in[6];
    const float* wl     = (const float*)d_in[7];
    float* out = (float*)d_out;

    float*  opd   = (float*)d_ws;
    float2* field = (float2*)(opd + (size_t)CHUNK * GG);
    float*  psf   = (float*)(field + (size_t)IMGS * GG);

    for (int arm = 0; arm < 2; ++arm) {
        const float* zern = arm ? zern90 : zern0;
        const float* foc  = arm ? foc90  : foc0;
        for (int base = 0; base < BATCH; base += CHUNK) {
            gemm_opd_wmma<<<dim3(GG / 64), 128, 0, stream>>>(opd, basis, zern, d_obj, foc, base);
            rowfft      <<<dim3(8, IMGS), 256, 0, stream>>>(field, opd, aper, wl);
            colfft      <<<dim3(8, IMGS), 256, 0, stream>>>(psf, field);
            sample_norm <<<dim3(IMGS),    256, 0, stream>>>(out, psf, wl, arm, base);
        }
    }
}